// LSTMClassifier_88983132438778
// MI455X (gfx1250) — compile-verified
//
#include <hip/hip_runtime.h>

namespace {

constexpr int T_SEQ = 336;
constexpr int I_DIM = 9;
constexpr int H_DIM = 64;
constexpr int MT    = 32;     // batch rows per workgroup
constexpr int NTH   = 256;    // 8 x wave32
constexpr int KS0   = 3;      // K-steps (K=4 each) for layer-0 input proj (9 padded to 12)
constexpr int KSH   = 16;     // K-steps for K=64
constexpr int KS1   = 32;     // K-steps for layer-1 fused K=128 ([w_ih1 ; w_hh1])
constexpr int HP    = 66;     // LDS pitch for h tiles (even -> 8B-aligned float2)
constexpr int XP    = 14;     // LDS pitch for x tile

constexpr int SZ_WIH0 = 16 * KS0 * 64;   // 3072 floats
constexpr int SZ_WHH0 = 16 * KSH * 64;   // 16384 floats
constexpr int SZ_W1   = 16 * KS1 * 64;   // 32768 floats
constexpr int SZ_H    = MT * HP;         // 2112 floats
constexpr int SZ_X    = MT * XP;         // 448 floats

constexpr int OFF_WIH0 = 0;
constexpr int OFF_WHH0 = OFF_WIH0 + SZ_WIH0;
constexpr int OFF_W1   = OFF_WHH0 + SZ_WHH0;
constexpr int OFF_H0   = OFF_W1 + SZ_W1;
constexpr int OFF_H1   = OFF_H0 + SZ_H;
constexpr int OFF_X    = OFF_H1 + SZ_H;
constexpr int SM_TOT   = OFF_X + SZ_X;   // 56896 floats = 222.3 KB (< 320 KB/WG)

typedef float v2f __attribute__((ext_vector_type(2)));
typedef float v8f __attribute__((ext_vector_type(8)));

__device__ __forceinline__ v8f wmma4(v2f a, v2f b, v8f c) {
  // D = A(16x4,f32) * B(4x16,f32) + C(16x16,f32)
  return __builtin_amdgcn_wmma_f32_16x16x4_f32(false, a, false, b, (short)0, c, false, false);
}

// gfx1250 has a hardware v_tanh_f32 (TRANS pipe). Build the LSTM gate
// nonlinearities on it: tanh = 1 op, sigmoid(x) = 0.5*tanh(0.5x)+0.5 = 1 TRANS + 1 fma.
#if __has_builtin(__builtin_amdgcn_tanhf)
__device__ __forceinline__ float tanh_(float x)    { return __builtin_amdgcn_tanhf(x); }
__device__ __forceinline__ float sigmoid_(float x) {
  return __builtin_fmaf(0.5f, __builtin_amdgcn_tanhf(0.5f * x), 0.5f);
}
#else
// fallback: exp + single-instruction v_rcp_f32 (avoid IEEE divide expansion)
__device__ __forceinline__ float sigmoid_(float x) {
  return __builtin_amdgcn_rcpf(1.0f + __expf(-x));
}
__device__ __forceinline__ float tanh_(float x) {
  return 1.0f - 2.0f * __builtin_amdgcn_rcpf(1.0f + __expf(2.0f * x));
}
#endif

__global__ __launch_bounds__(NTH, 1)
void lstm2_fused_kernel(const float* __restrict__ x,
                        const float* __restrict__ w_ih0, const float* __restrict__ w_hh0,
                        const float* __restrict__ b_ih0, const float* __restrict__ b_hh0,
                        const float* __restrict__ w_ih1, const float* __restrict__ w_hh1,
                        const float* __restrict__ b_ih1, const float* __restrict__ b_hh1,
                        const float* __restrict__ w_lin, const float* __restrict__ b_lin,
                        float* __restrict__ out)
{
  __shared__ float smem[SM_TOT];
  float* __restrict__ wih0f = smem + OFF_WIH0;
  float* __restrict__ whh0f = smem + OFF_WHH0;
  float* __restrict__ w1f   = smem + OFF_W1;
  float* __restrict__ h0l   = smem + OFF_H0;
  float* __restrict__ h1l   = smem + OFF_H1;
  float* __restrict__ xl    = smem + OFF_X;

  const int tid  = threadIdx.x;
  const int lane = tid & 31;
  const int wv   = tid >> 5;
  const int mh   = wv >> 2;      // which 16-row half of the 32-row slab
  const int jq   = wv & 3;       // which 16-wide hidden-column quarter
  const int l15  = lane & 15;
  const int hf   = lane >> 4;
  const int mbase = mh * 16;
  const int row0 = blockIdx.x * MT;

  // ---- stage weights into LDS in exact WMMA B-fragment layout -------------
  // fragment index F = (((nt*KS + kk)*32) + lane)*2 + v  maps to B[k][n] = W[n][k]
  for (int F = tid; F < SZ_WIH0; F += NTH) {
    const int v = F & 1, l = (F >> 1) & 31;
    const int kk = (F >> 6) % KS0, nt = F / (KS0 * 64);
    const int n = nt * 16 + (l & 15);
    const int k = kk * 4 + (l >> 4) * 2 + v;
    wih0f[F] = (k < I_DIM) ? w_ih0[n * I_DIM + k] : 0.0f;
  }
  for (int F = tid; F < SZ_WHH0; F += NTH) {
    const int v = F & 1, l = (F >> 1) & 31;
    const int kk = (F >> 6) % KSH, nt = F / (KSH * 64);
    const int n = nt * 16 + (l & 15);
    const int k = kk * 4 + (l >> 4) * 2 + v;
    whh0f[F] = w_hh0[n * H_DIM + k];
  }
  for (int F = tid; F < SZ_W1; F += NTH) {
    const int v = F & 1, l = (F >> 1) & 31;
    const int kk = (F >> 6) % KS1, nt = F / (KS1 * 64);
    const int n = nt * 16 + (l & 15);
    const int k = kk * 4 + (l >> 4) * 2 + v;          // 0..127 across [w_ih1 ; w_hh1]
    w1f[F] = (k < H_DIM) ? w_ih1[n * H_DIM + k] : w_hh1[n * H_DIM + (k - H_DIM)];
  }
  for (int F = tid; F < SZ_H; F += NTH) { h0l[F] = 0.0f; h1l[F] = 0.0f; }

  // per-lane gate biases (D-fragment column n = lane&15 within the tile)
  float bias0[4], bias1[4];
  #pragma unroll
  for (int g = 0; g < 4; ++g) {
    const int n = g * 64 + jq * 16 + l15;
    bias0[g] = b_ih0[n] + b_hh0[n];
    bias1[g] = b_ih1[n] + b_hh1[n];
  }
  float c0r[8], c1r[8];
  #pragma unroll
  for (int v = 0; v < 8; ++v) { c0r[v] = 0.0f; c1r[v] = 0.0f; }

  for (int t = 0; t < T_SEQ; ++t) {
    // stage x_t slab (32 rows x 9, zero-padded to K=12)
    for (int F = tid; F < MT * 12; F += NTH) {
      const int m = F / 12, k = F % 12;
      xl[m * XP + k] = (k < I_DIM) ? x[(row0 + m) * (T_SEQ * I_DIM) + t * I_DIM + k] : 0.0f;
    }
    __syncthreads();   // x_t ready; previous step's h1 visible

    // ---- layer 0: gates = bias + x_t @ Wih0^T + h0 @ Whh0^T ----
    v8f acc[4];
    #pragma unroll
    for (int g = 0; g < 4; ++g) {
      v8f z;
      #pragma unroll
      for (int v = 0; v < 8; ++v) z[v] = bias0[g];
      acc[g] = z;
    }
    {
      const float* Ax = xl + (mbase + l15) * XP + hf * 2;
      #pragma unroll
      for (int kk = 0; kk < KS0; ++kk) {
        const v2f a = *(const v2f*)(Ax + kk * 4);
        #pragma unroll
        for (int g = 0; g < 4; ++g) {
          const v2f b = *(const v2f*)(wih0f + ((g * 4 + jq) * KS0 + kk) * 64 + 2 * lane);
          acc[g] = wmma4(a, b, acc[g]);
        }
      }
      const float* Ah = h0l + (mbase + l15) * HP + hf * 2;
      #pragma unroll
      for (int kk = 0; kk < KSH; ++kk) {
        const v2f a = *(const v2f*)(Ah + kk * 4);
        #pragma unroll
        for (int g = 0; g < 4; ++g) {
          const v2f b = *(const v2f*)(whh0f + ((g * 4 + jq) * KSH + kk) * 64 + 2 * lane);
          acc[g] = wmma4(a, b, acc[g]);
        }
      }
    }
    float hnew[8];
    #pragma unroll
    for (int v = 0; v < 8; ++v) {
      const float ig = sigmoid_(acc[0][v]);
      const float fg = sigmoid_(acc[1][v]);
      const float gg = tanh_(acc[2][v]);
      const float og = sigmoid_(acc[3][v]);
      const float c  = fg * c0r[v] + ig * gg;
      c0r[v] = c;
      hnew[v] = og * tanh_(c);
    }
    __syncthreads();   // all waves finished reading h0l
    #pragma unroll
    for (int v = 0; v < 8; ++v)
      h0l[(mbase + v + 8 * hf) * HP + jq * 16 + l15] = hnew[v];
    __syncthreads();   // new h0 visible

    // ---- layer 1: gates = bias + [h0 ; h1] @ [Wih1 ; Whh1]^T  (K = 128) ----
    #pragma unroll
    for (int g = 0; g < 4; ++g) {
      v8f z;
      #pragma unroll
      for (int v = 0; v < 8; ++v) z[v] = bias1[g];
      acc[g] = z;
    }
    {
      const float* Ah0 = h0l + (mbase + l15) * HP + hf * 2;
      #pragma unroll
      for (int kk = 0; kk < KSH; ++kk) {
        const v2f a = *(const v2f*)(Ah0 + kk * 4);
        #pragma unroll
        for (int g = 0; g < 4; ++g) {
          const v2f b = *(const v2f*)(w1f + ((g * 4 + jq) * KS1 + kk) * 64 + 2 * lane);
          acc[g] = wmma4(a, b, acc[g]);
        }
      }
      const float* Ah1 = h1l + (mbase + l15) * HP + hf * 2;
      #pragma unroll
      for (int kk = 0; kk < KSH; ++kk) {
        const v2f a = *(const v2f*)(Ah1 + kk * 4);
        #pragma unroll
        for (int g = 0; g < 4; ++g) {
          const v2f b = *(const v2f*)(w1f + ((g * 4 + jq) * KS1 + (KSH + kk)) * 64 + 2 * lane);
          acc[g] = wmma4(a, b, acc[g]);
        }
      }
    }
    #pragma unroll
    for (int v = 0; v < 8; ++v) {
      const float ig = sigmoid_(acc[0][v]);
      const float fg = sigmoid_(acc[1][v]);
      const float gg = tanh_(acc[2][v]);
      const float og = sigmoid_(acc[3][v]);
      const float c  = fg * c1r[v] + ig * gg;
      c1r[v] = c;
      hnew[v] = og * tanh_(c);
    }
    __syncthreads();   // all waves finished reading h1l
    #pragma unroll
    for (int v = 0; v < 8; ++v)
      h1l[(mbase + v + 8 * hf) * HP + jq * 16 + l15] = hnew[v];
    // next iteration's top barrier publishes h1 before it is read
  }
  __syncthreads();

  // ---- head: out = sigmoid(h1_last @ w_lin^T + b_lin) ----
  if (tid < MT) {
    float s = b_lin[0];
    #pragma unroll
    for (int k = 0; k < H_DIM; ++k) s += h1l[tid * HP + k] * w_lin[k];
    out[row0 + tid] = sigmoid_(s);
  }
}

} // namespace

extern "C" void kernel_launch(void* const* d_in, const int* in_sizes, int n_in,
                              void* d_out, int out_size, void* d_ws, size_t ws_size,
                              hipStream_t stream) {
  const float* x     = (const float*)d_in[0];
  const float* w_ih0 = (const float*)d_in[1];
  const float* w_hh0 = (const float*)d_in[2];
  const float* b_ih0 = (const float*)d_in[3];
  const float* b_hh0 = (const float*)d_in[4];
  const float* w_ih1 = (const float*)d_in[5];
  const float* w_hh1 = (const float*)d_in[6];
  const float* b_ih1 = (const float*)d_in[7];
  const float* b_hh1 = (const float*)d_in[8];
  const float* w_lin = (const float*)d_in[9];
  const float* b_lin = (const float*)d_in[10];
  float* out = (float*)d_out;

  const int batch  = in_sizes[0] / (T_SEQ * I_DIM);  // 4096
  const int blocks = batch / MT;                     // 128 workgroups x 256 threads
  lstm2_fused_kernel<<<blocks, NTH, 0, stream>>>(x, w_ih0, w_hh0, b_ih0, b_hh0,
                                                 w_ih1, w_hh1, b_ih1, b_hh1,
                                                 w_lin, b_lin, out);
}